// VariationalGraphDecoder_83519934038614
// MI455X (gfx1250) — compile-verified
//
#include <hip/hip_runtime.h>

typedef __attribute__((ext_vector_type(2))) float v2f;
typedef __attribute__((ext_vector_type(8))) float v8f;

#define CIN 128   // in/hidden channels (K of both GEMMs)

// ---------------------------------------------------------------- utilities
__global__ void copy_f32v4(float4* __restrict__ dst, const float4* __restrict__ src, long n4) {
  long i = (long)blockIdx.x * blockDim.x + threadIdx.x;
  if (i < n4) dst[i] = src[i];
}

// dst[perm[r]*CIN + c] += (relu? max(src,0) : src)[r*CIN + c]; perm entries unique -> plain add
__global__ void scatter_add_rows(float* __restrict__ dst, const float* __restrict__ src,
                                 const int* __restrict__ perm, int nrows, int do_relu) {
  long i = (long)blockIdx.x * blockDim.x + threadIdx.x;
  long total = (long)nrows * CIN;
  if (i >= total) return;
  int r = (int)(i >> 7);          // /128
  int c = (int)(i & 127);
  float v = src[i];
  if (do_relu) v = v > 0.0f ? v : 0.0f;
  dst[(long)perm[r] * CIN + c] += v;
}

// ---------------------------------------------------------------- degree / norm
__global__ void deg_init(float* __restrict__ deg, int n) {
  int i = blockIdx.x * blockDim.x + threadIdx.x;
  if (i < n) deg[i] = 2.0f;       // improved=True self-loop weight
}

__global__ void deg_edges(float* __restrict__ deg, const int* __restrict__ col,
                          const float* __restrict__ w, int E) {
  int e = blockIdx.x * blockDim.x + threadIdx.x;
  if (e < E) atomicAdd(&deg[col[e]], w[e]);
}

__global__ void deg_rsqrt(float* __restrict__ deg, int n) {
  int i = blockIdx.x * blockDim.x + threadIdx.x;
  if (i < n) { float d = deg[i]; deg[i] = d > 0.0f ? rsqrtf(d) : 0.0f; }
}

// ---------------------------------------------------------------- WMMA GEMM
// out[M x NCOLS] = A[M x 128] @ W[128 x NCOLS], fp32 via V_WMMA_F32_16X16X4_F32.
//  - W staged once per block into LDS in fragment-swizzled layout:
//      sB[((nt*32 + ks)*2 + half)*16 + ncol] = { W[(4ks+2h)*N + nt*16+ncol],
//                                               W[(4ks+2h+1)*N + nt*16+ncol] }
//    -> each B fragment is ONE ds_load_b64, lanes 0-15 hit 16 consecutive
//       float2s (conflict-free across the 64 banks).
//  - One wave per 16-row M tile; 16x128 A panel cached in 64 VGPRs and reused
//    across all NCOLS/16 output tiles.
//  - Per N-tile: DS clause preloads all 32 B frags, then 32 back-to-back WMMAs
//    on two interleaved accumulators (breaks the serial D->C chain).
// M must be a multiple of 16 (50000 and 100000 both are).
template <int NCOLS>
__global__ void __launch_bounds__(256) gemm_wmma_f32(const float* __restrict__ A,
                                                     const float* __restrict__ W,
                                                     float* __restrict__ out, int Mtiles) {
  __shared__ float2 sB[(size_t)NCOLS * 64];   // NCOLS*128 floats (64KB / 32KB)

  const int tid = threadIdx.x;
  // ---- cooperative fragment-swizzled staging of W ----
  for (int idx = tid; idx < NCOLS * 64; idx += 256) {
    int ncol = idx & 15;
    int h    = (idx >> 4) & 1;
    int ks   = (idx >> 5) & 31;
    int nt   = idx >> 10;
    int k0   = ks * 4 + h * 2;
    int col  = nt * 16 + ncol;
    float2 t;
    t.x = W[(long)k0 * NCOLS + col];
    t.y = W[(long)(k0 + 1) * NCOLS + col];
    sB[idx] = t;
  }
  __syncthreads();

  const int wid  = blockIdx.x * 8 + (tid >> 5);
  const int lane = tid & 31;
  if (wid >= Mtiles) return;

  const int rowBase = wid << 4;
  const int half    = lane >> 4;        // 0: lanes 0-15, 1: lanes 16-31
  const int mrow    = lane & 15;        // A: M index of this lane
  const int koff    = half << 1;        // A/B: K offset within 4-wide step
  const int ncol    = lane & 15;        // B/D: N index of this lane
  const int mBase   = half << 3;        // D: M base of this lane half

  // Cache the whole 16x128 A panel: 32 k-steps x v2f = 64 VGPRs.
  v2f a[32];
  const float* arow = A + (long)(rowBase + mrow) * CIN + koff;
#pragma unroll
  for (int ks = 0; ks < 32; ++ks) {
    a[ks].x = arow[ks * 4];
    a[ks].y = arow[ks * 4 + 1];
  }

#pragma unroll
  for (int nt = 0; nt < NCOLS / 16; ++nt) {
    // preload all 32 B fragments for this N-tile from LDS (one ds clause)
    v2f b[32];
#pragma unroll
    for (int ks = 0; ks < 32; ++ks) {
      float2 t = sB[(((nt * 32 + ks) * 2 + half) << 4) + ncol];
      b[ks].x = t.x;
      b[ks].y = t.y;
    }
    // 32 back-to-back WMMAs, two interleaved accumulator chains
    v8f acc0 = {}, acc1 = {};
#pragma unroll
    for (int ks = 0; ks < 32; ks += 2) {
      acc0 = __builtin_amdgcn_wmma_f32_16x16x4_f32(false, a[ks],     false, b[ks],
                                                   (short)0, acc0, false, false);
      acc1 = __builtin_amdgcn_wmma_f32_16x16x4_f32(false, a[ks + 1], false, b[ks + 1],
                                                   (short)0, acc1, false, false);
    }
    v8f acc = acc0 + acc1;
    const int colBase = nt << 4;
#pragma unroll
    for (int i = 0; i < 8; ++i)
      out[(long)(rowBase + mBase + i) * NCOLS + colBase + ncol] = acc[i];
  }
}

// ---------------------------------------------------------------- aggregation
// acc[j,c] = bias[c] + (dinv[j]*2*dinv[j]) * xw[j,c]   (self-loop + bias init)
template <int NCOLS>
__global__ void out_init(float* __restrict__ acc, const float* __restrict__ xw,
                         const float* __restrict__ dinv, const float* __restrict__ bias,
                         int nrows) {
  long i = (long)blockIdx.x * blockDim.x + threadIdx.x;
  long total = (long)nrows * NCOLS;
  if (i >= total) return;
  int r = (int)(i / NCOLS);
  int c = (int)(i % NCOLS);
  float d = dinv[r];
  acc[i] = bias[c] + 2.0f * d * d * xw[i];
}

// One wave per edge: acc[col,:] += (dinv[row]*w*dinv[col]) * xw[row,:]
template <int NCOLS>
__global__ void edge_scatter(float* __restrict__ acc, const float* __restrict__ xw,
                             const int* __restrict__ ei, const float* __restrict__ w,
                             const float* __restrict__ dinv, int E) {
  long t = (long)blockIdx.x * blockDim.x + threadIdx.x;
  int e = (int)(t >> 5);
  int lane = threadIdx.x & 31;
  if (e >= E) return;
  int row = ei[e];
  int col = ei[(long)E + e];
  float coef = dinv[row] * w[e] * dinv[col];
  constexpr int V = NCOLS / 32;                 // 4 (C=128) or 2 (C=64)
  const float* xr = xw + (long)row * NCOLS + lane * V;
  float* dr = acc + (long)col * NCOLS + lane * V;
#pragma unroll
  for (int v = 0; v < V; ++v) atomicAdd(&dr[v], coef * xr[v]);
}

// ---------------------------------------------------------------- launch
static inline int cdiv(long a, long b) { return (int)((a + b - 1) / b); }

extern "C" void kernel_launch(void* const* d_in, const int* in_sizes, int n_in,
                              void* d_out, int out_size, void* d_ws, size_t ws_size,
                              hipStream_t stream) {
  const float* x     = (const float*)d_in[0];
  const float* xs0   = (const float*)d_in[1];
  const float* xs1   = (const float*)d_in[2];
  const int*   ei0   = (const int*)d_in[3];
  const int*   ei1   = (const int*)d_in[4];
  const float* we0   = (const float*)d_in[5];
  const float* we1   = (const float*)d_in[6];
  const int*   perm0 = (const int*)d_in[7];
  const int*   perm1 = (const int*)d_in[8];
  const float* W0    = (const float*)d_in[9];
  const float* b0    = (const float*)d_in[10];
  const float* W1    = (const float*)d_in[11];
  const float* b1    = (const float*)d_in[12];

  const int N2 = in_sizes[0] / CIN;   // 25000
  const int N0 = in_sizes[1] / CIN;   // 100000
  const int N1 = in_sizes[2] / CIN;   // 50000
  const int E0 = in_sizes[5];         // 1600000
  const int E1 = in_sizes[6];         // 800000

  // workspace layout (floats)
  float* ws    = (float*)d_ws;
  float* hin1  = ws;                               // N1*128
  float* xw1   = hin1 + (size_t)N1 * CIN;          // N1*128
  float* acc1  = xw1  + (size_t)N1 * CIN;          // N1*128
  float* hin0  = acc1 + (size_t)N1 * CIN;          // N0*128
  float* xw0   = hin0 + (size_t)N0 * CIN;          // N0*64
  float* dinv1 = xw0  + (size_t)N0 * 64;           // N1
  float* dinv0 = dinv1 + N1;                       // N0
  float* out   = (float*)d_out;                    // N0*64

  // ================= layer 1: coarse -> mid (128 -> 128, ReLU) =================
  copy_f32v4<<<cdiv((long)N1 * CIN / 4, 256), 256, 0, stream>>>(
      (float4*)hin1, (const float4*)xs1, (long)N1 * CIN / 4);
  scatter_add_rows<<<cdiv((long)N2 * CIN, 256), 256, 0, stream>>>(hin1, x, perm1, N2, 0);

  deg_init <<<cdiv(N1, 256), 256, 0, stream>>>(dinv1, N1);
  deg_edges<<<cdiv(E1, 256), 256, 0, stream>>>(dinv1, ei1 + E1, we1, E1);
  deg_rsqrt<<<cdiv(N1, 256), 256, 0, stream>>>(dinv1, N1);

  gemm_wmma_f32<128><<<cdiv(N1 / 16, 8), 256, 0, stream>>>(hin1, W0, xw1, N1 / 16);

  out_init<128><<<cdiv((long)N1 * 128, 256), 256, 0, stream>>>(acc1, xw1, dinv1, b0, N1);
  edge_scatter<128><<<cdiv((long)E1 * 32, 256), 256, 0, stream>>>(acc1, xw1, ei1, we1, dinv1, E1);

  // ================= layer 0: mid -> fine (128 -> 64, no act) ==================
  copy_f32v4<<<cdiv((long)N0 * CIN / 4, 256), 256, 0, stream>>>(
      (float4*)hin0, (const float4*)xs0, (long)N0 * CIN / 4);
  scatter_add_rows<<<cdiv((long)N1 * CIN, 256), 256, 0, stream>>>(hin0, acc1, perm0, N1, 1);

  deg_init <<<cdiv(N0, 256), 256, 0, stream>>>(dinv0, N0);
  deg_edges<<<cdiv(E0, 256), 256, 0, stream>>>(dinv0, ei0 + E0, we0, E0);
  deg_rsqrt<<<cdiv(N0, 256), 256, 0, stream>>>(dinv0, N0);

  gemm_wmma_f32<64><<<cdiv(N0 / 16, 8), 256, 0, stream>>>(hin0, W1, xw0, N0 / 16);

  out_init<64><<<cdiv((long)N0 * 64, 256), 256, 0, stream>>>(out, xw0, dinv0, b1, N0);
  edge_scatter<64><<<cdiv((long)E0 * 32, 256), 256, 0, stream>>>(out, xw0, ei0, we0, dinv0, E0);
}